// TTT_layer_74302934221150
// MI455X (gfx1250) — compile-verified
//
#include <hip/hip_runtime.h>

// ---------------------------------------------------------------------------
// TTT-linear layer for MI455X (gfx1250), f32 WMMA (V_WMMA_F32_16X16X4_F32).
// Chunked dual-form: removes the token-serial matvec, feature dim decouples
// so the sequential part runs sync-free across (batch x e-tile) workgroups.
// ---------------------------------------------------------------------------

#define N_BATCH 8
#define S_LEN   2048
#define D_DIM   1024
#define CHUNK   64
#define EW      64          // e-columns owned per solver workgroup
#define LR_C    0.01f

typedef __attribute__((ext_vector_type(2))) float v2f;
typedef __attribute__((ext_vector_type(8))) float v8f;

__device__ __forceinline__ v8f wmma_f32(v2f a, v2f b, v8f c) {
  // D = A(16x4) * B(4x16) + C, f32 in/out, wave32.
  return __builtin_amdgcn_wmma_f32_16x16x4_f32(false, a, false, b, (short)0, c,
                                               false, false);
}

// ---------------------------------------------------------------------------
// Stage 1: C[M,N] = A[M,K] @ B[N,K]^T  (row-major, K contiguous both sides)
// 128x128x32 tiles, 256 threads = 8 waves (2x4), each wave 64x32 = 4x2 WMMA
// tiles. LDS-staged slabs, padded to 36 floats/row (16B-aligned, no 64-bank
// conflicts on fragment reads).
// ---------------------------------------------------------------------------
#define BM  128
#define BN  128
#define BK  32
#define PAD 36

__global__ __launch_bounds__(256) void gemm_nt(
    const float* __restrict__ A, size_t strideA,
    const float* __restrict__ B, size_t strideB,
    float* __restrict__ C, size_t strideC,
    const int Kdim, const int Ndim) {
  __shared__ float As[BM * PAD];
  __shared__ float Bs[BN * PAD];
  const int zb = blockIdx.z;
  A += (size_t)zb * strideA;
  B += (size_t)zb * strideB;
  C += (size_t)zb * strideC;
  const int m0 = blockIdx.y * BM, n0 = blockIdx.x * BN;
  const int tid = threadIdx.x, lane = tid & 31, wid = tid >> 5;
  const int lo = lane & 15, hi = lane >> 4;
  const int mw = (wid >> 2) * 64, nw = (wid & 3) * 32;

  v8f acc[4][2];
  for (int i = 0; i < 4; ++i)
    for (int j = 0; j < 2; ++j) { v8f z = {}; acc[i][j] = z; }

  for (int kb = 0; kb < Kdim; kb += BK) {
    // 1024 float4 per slab, 4 per thread, coalesced along K.
    for (int i = 0; i < 4; ++i) {
      const int flat = i * 256 + tid;
      const int r = flat >> 3, c4 = (flat & 7) << 2;
      *(float4*)&As[r * PAD + c4] =
          *(const float4*)&A[(size_t)(m0 + r) * Kdim + kb + c4];
      *(float4*)&Bs[r * PAD + c4] =
          *(const float4*)&B[(size_t)(n0 + r) * Kdim + kb + c4];
    }
    __syncthreads();
#pragma unroll
    for (int kk = 0; kk < BK; kk += 4) {
      v2f af[4], bf[2];
      for (int i = 0; i < 4; ++i)
        af[i] = *(const v2f*)&As[(mw + i * 16 + lo) * PAD + kk + 2 * hi];
      for (int j = 0; j < 2; ++j)
        bf[j] = *(const v2f*)&Bs[(nw + j * 16 + lo) * PAD + kk + 2 * hi];
      for (int i = 0; i < 4; ++i)
        for (int j = 0; j < 2; ++j) acc[i][j] = wmma_f32(af[i], bf[j], acc[i][j]);
    }
    __syncthreads();
  }
  for (int i = 0; i < 4; ++i)
    for (int j = 0; j < 2; ++j)
      for (int v = 0; v < 8; ++v)
        C[(size_t)(m0 + mw + i * 16 + v + 8 * hi) * Ndim + n0 + nw + j * 16 + lo] =
            acc[i][j][v];
}

// U = c0 * (bK + b0 - V), in place over the bK buffer. Exact grid, no tail.
__global__ void make_u(float* __restrict__ U, const float* __restrict__ V,
                       const float* __restrict__ b0, const float c0) {
  const size_t i = (size_t)blockIdx.x * blockDim.x + threadIdx.x;
  const int d = (int)(i & (D_DIM - 1));
  U[i] = c0 * (U[i] + b0[d] - V[i]);
}

// ---------------------------------------------------------------------------
// Stage 2: sequential chunk solver. Grid (D/EW, B); each WG owns e-range
// [E0, E0+EW) of Wt rows / bt / output columns for one batch -> no inter-WG
// synchronization. 256 threads = 8 waves. LDS: G chunk (64xEW) + Akk + Akq.
// ---------------------------------------------------------------------------
__global__ __launch_bounds__(256) void ttt_solver(
    const float* __restrict__ Kbuf, const float* __restrict__ Qbuf,
    const float* __restrict__ Ubuf, const float* __restrict__ b0,
    float* __restrict__ Wt_all, float* __restrict__ bt_all,
    float* __restrict__ out_all) {
  __shared__ float Gs[CHUNK * EW];       // 16 KB: g_t for this chunk, cols E0..
  __shared__ float Akk[CHUNK * CHUNK];   // 16 KB: k_t . k_i + 1
  __shared__ float Akq[CHUNK * CHUNK];   // 16 KB: k_i . q_t + 1
  const int bat = blockIdx.y;
  const int E0 = blockIdx.x * EW;
  const float* Kb = Kbuf + (size_t)bat * S_LEN * D_DIM;
  const float* Qb = Qbuf + (size_t)bat * S_LEN * D_DIM;
  const float* Ub = Ubuf + (size_t)bat * S_LEN * D_DIM;
  float* Wt = Wt_all + (size_t)bat * D_DIM * D_DIM;  // Wt[e][d] = sum g_i[e] k_i[d]
  float* bt = bt_all + (size_t)bat * D_DIM;          // bt[e]    = sum g_i[e]
  float* outb = out_all + (size_t)bat * S_LEN * D_DIM;
  const int tid = threadIdx.x, lane = tid & 31, wid = tid >> 5;
  const int lo = lane & 15, hi = lane >> 4;
  const float c1 = (2.0f / (float)D_DIM) * LR_C;

  for (int c = 0; c < S_LEN / CHUNK; ++c) {
    const int t0 = c * CHUNK;
    const float* Kc = Kb + (size_t)t0 * D_DIM;
    const float* Qc = Qb + (size_t)t0 * D_DIM;

    // P1: Akk[t][i] and Akq[i][t] (full-D dot products, WMMA, 32 tiles).
    for (int tidx = wid; tidx < 32; tidx += 8) {
      const int which = tidx >> 4;  // 0 -> Akk, 1 -> Akq
      const int mi = (tidx >> 2) & 3, ni = tidx & 3;
      const float* ar = Kc + (size_t)(mi * 16 + lo) * D_DIM + 2 * hi;
      const float* br = (which ? Qc : Kc) + (size_t)(ni * 16 + lo) * D_DIM + 2 * hi;
      v8f acc = {};
#pragma unroll 8
      for (int k = 0; k < D_DIM; k += 4)
        acc = wmma_f32(*(const v2f*)(ar + k), *(const v2f*)(br + k), acc);
      float* dst = which ? Akq : Akk;
      for (int v = 0; v < 8; ++v)
        dst[(mi * 16 + v + 8 * hi) * CHUNK + ni * 16 + lo] = acc[v] + 1.0f;
    }
    // P2: G init = U - c1 * (Kc @ Wt^T + bt)   (cross-chunk term, WMMA).
    for (int tidx = wid; tidx < 16; tidx += 8) {
      const int mi = tidx >> 2, ni = tidx & 3;
      const float* ar = Kc + (size_t)(mi * 16 + lo) * D_DIM + 2 * hi;
      const float* br = Wt + (size_t)(E0 + ni * 16 + lo) * D_DIM + 2 * hi;
      v8f acc = {};
#pragma unroll 8
      for (int k = 0; k < D_DIM; k += 4)
        acc = wmma_f32(*(const v2f*)(ar + k), *(const v2f*)(br + k), acc);
      for (int v = 0; v < 8; ++v) {
        const int r = mi * 16 + v + 8 * hi, col = ni * 16 + lo;
        Gs[r * EW + col] =
            Ub[(size_t)(t0 + r) * D_DIM + E0 + col] - c1 * (acc[v] + bt[E0 + col]);
      }
    }
    __syncthreads();
    // P3: forward substitution, sequential over t within the chunk.
    for (int t = 1; t < CHUNK; ++t) {
      if (tid < EW) {
        float s = 0.f;
        for (int i = 0; i < t; ++i) s += Akk[t * CHUNK + i] * Gs[i * EW + tid];
        Gs[t * EW + tid] -= c1 * s;
      }
      __syncthreads();
    }
    // P4: out = bQ + b0 - LR * (Qc @ Wt^T + bt + sum_{i<=t} Akq[i][t] g_i).
    for (int tidx = wid; tidx < 16; tidx += 8) {
      const int mi = tidx >> 2, ni = tidx & 3;
      const float* ar = Qc + (size_t)(mi * 16 + lo) * D_DIM + 2 * hi;
      const float* br = Wt + (size_t)(E0 + ni * 16 + lo) * D_DIM + 2 * hi;
      v8f acc = {};
#pragma unroll 8
      for (int k = 0; k < D_DIM; k += 4)
        acc = wmma_f32(*(const v2f*)(ar + k), *(const v2f*)(br + k), acc);
      for (int v = 0; v < 8; ++v) {
        const int r = mi * 16 + v + 8 * hi, col = ni * 16 + lo;
        float s = 0.f;
        for (int i = 0; i <= r; ++i) s += Akq[i * CHUNK + r] * Gs[i * EW + col];
        const size_t o = (size_t)(t0 + r) * D_DIM + E0 + col;
        outb[o] = outb[o] + b0[E0 + col] - LR_C * (acc[v] + bt[E0 + col] + s);
      }
    }
    __syncthreads();  // Wt reads (P2/P4) complete before P5 writes Wt
    // P5: Wt[E0..E0+EW) += Gs^T @ Kc  (rank-64 update, WMMA); bt += colsum(Gs).
    for (int tidx = wid; tidx < (EW / 16) * (D_DIM / 16); tidx += 8) {
      const int mi = tidx & 3, ni = tidx >> 2;
      float* wp = Wt + (size_t)(E0 + mi * 16) * D_DIM + ni * 16;
      v8f acc;
      for (int v = 0; v < 8; ++v) acc[v] = wp[(size_t)(v + 8 * hi) * D_DIM + lo];
#pragma unroll
      for (int kk = 0; kk < CHUNK; kk += 4) {
        const int kr = kk + 2 * hi;
        v2f a, b;
        a.x = Gs[kr * EW + mi * 16 + lo];
        a.y = Gs[(kr + 1) * EW + mi * 16 + lo];
        b.x = Kc[(size_t)kr * D_DIM + ni * 16 + lo];
        b.y = Kc[(size_t)(kr + 1) * D_DIM + ni * 16 + lo];
        acc = wmma_f32(a, b, acc);
      }
      for (int v = 0; v < 8; ++v) wp[(size_t)(v + 8 * hi) * D_DIM + lo] = acc[v];
    }
    if (tid < EW) {
      float s = 0.f;
      for (int i = 0; i < CHUNK; ++i) s += Gs[i * EW + tid];
      bt[E0 + tid] += s;
    }
    __threadfence_block();
    __syncthreads();  // Wt/bt/Gs settled before next chunk
  }
}

// ---------------------------------------------------------------------------
extern "C" void kernel_launch(void* const* d_in, const int* in_sizes, int n_in,
                              void* d_out, int out_size, void* d_ws, size_t ws_size,
                              hipStream_t stream) {
  (void)in_sizes; (void)n_in; (void)out_size; (void)ws_size;
  const float* x   = (const float*)d_in[0];   // in_seq  [B,S,D]
  const float* thK = (const float*)d_in[1];   // theta_K [D,D]
  const float* thV = (const float*)d_in[2];   // theta_V [D,D]
  const float* thQ = (const float*)d_in[3];   // theta_Q [D,D]
  const float* W0  = (const float*)d_in[4];   // W0      [D,D]
  const float* b0  = (const float*)d_in[5];   // b0      [D]
  float* out = (float*)d_out;                 // [B,S,D]

  const size_t nBSD = (size_t)N_BATCH * S_LEN * D_DIM;
  float* Kbuf = (float*)d_ws;
  float* Vbuf = Kbuf + nBSD;
  float* Qbuf = Vbuf + nBSD;
  float* Ubuf = Qbuf + nBSD;                          // bK, then U in place
  float* Wt   = Ubuf + nBSD;                          // [B][D][D]
  float* bt   = Wt + (size_t)N_BATCH * D_DIM * D_DIM; // [B][D]

  const size_t sIn = (size_t)S_LEN * D_DIM;
  dim3 gproj(D_DIM / BN, S_LEN / BM, N_BATCH);
  gemm_nt<<<gproj, 256, 0, stream>>>(x, sIn, thK, 0, Kbuf, sIn, D_DIM, D_DIM);
  gemm_nt<<<gproj, 256, 0, stream>>>(x, sIn, thV, 0, Vbuf, sIn, D_DIM, D_DIM);
  gemm_nt<<<gproj, 256, 0, stream>>>(x, sIn, thQ, 0, Qbuf, sIn, D_DIM, D_DIM);
  gemm_nt<<<gproj, 256, 0, stream>>>(Kbuf, sIn, W0, 0, Ubuf, sIn, D_DIM, D_DIM);
  gemm_nt<<<gproj, 256, 0, stream>>>(Qbuf, sIn, W0, 0, out, sIn, D_DIM, D_DIM);
  make_u<<<(unsigned)(nBSD / 256), 256, 0, stream>>>(Ubuf, Vbuf, b0,
                                                     2.0f / (float)D_DIM);
  hipMemsetAsync(Wt, 0, (size_t)N_BATCH * D_DIM * D_DIM * sizeof(float), stream);
  hipMemsetAsync(bt, 0, (size_t)N_BATCH * D_DIM * sizeof(float), stream);
  dim3 gsol(D_DIM / EW, N_BATCH);
  ttt_solver<<<gsol, 256, 0, stream>>>(Kbuf, Qbuf, Ubuf, b0, Wt, bt, out);
}